// HeteroGNN_5677946765759
// MI455X (gfx1250) — compile-verified
//
#include <hip/hip_runtime.h>
#include <math.h>

typedef float v2f __attribute__((ext_vector_type(2)));
typedef float v8f __attribute__((ext_vector_type(8)));

#define HIDN 64
#define EDIMN 16

// ---------------------------------------------------------------------------
// C[M,N] = A[M,K] @ B[K,N] + bias[N]   (row-major, N multiple of 16, K mult 4)
// One wave32 computes one 16x16 tile via V_WMMA_F32_16X16X4_F32.
// A fragment (16x4): lane l -> row m=l&15, ks {kb,kb+1}, kb=(l>>4)*2 (2 VGPRs)
// B fragment (4x16): lane l -> col n=l&15, same k mapping (2 VGPRs)
// D (16x16, 8 VGPRs): vgpr v, lanes0-15 -> M=v ; lanes16-31 -> M=v+8
// ---------------------------------------------------------------------------
__global__ void gemm_bias_wmma(const float* __restrict__ A,
                               const float* __restrict__ B,
                               const float* __restrict__ bias,
                               float* __restrict__ C,
                               int M, int N, int K) {
  const int lane = threadIdx.x & 31;
  const int wid  = threadIdx.x >> 5;
  const int ntiles = N >> 4;
  const int mtiles = (M + 15) >> 4;
  int tile = blockIdx.x * (blockDim.x >> 5) + wid;
  if (tile >= ntiles * mtiles) return;
  const int mt = tile / ntiles;
  const int nt = tile - mt * ntiles;
  const int row0 = mt << 4;
  const int kb   = (lane >> 4) << 1;
  const int n    = (nt << 4) + (lane & 15);
  const int rowA = row0 + (lane & 15);
  const float okf = (rowA < M) ? 1.0f : 0.0f;      // branch-free row guard
  const int   rA  = (rowA < M) ? rowA : 0;
  const float* Arow = A + (size_t)rA * K;
  v8f acc = {0.f, 0.f, 0.f, 0.f, 0.f, 0.f, 0.f, 0.f};
  for (int k = 0; k < K; k += 4) {
    v2f a, b;
    a.x = Arow[k + kb]     * okf;
    a.y = Arow[k + kb + 1] * okf;
    b.x = B[(size_t)(k + kb)     * N + n];
    b.y = B[(size_t)(k + kb + 1) * N + n];
    acc = __builtin_amdgcn_wmma_f32_16x16x4_f32(false, a, false, b,
                                                (short)0, acc, false, false);
  }
  const float bv = bias ? bias[n] : 0.0f;
  const int rbase = row0 + ((lane >> 4) << 3);
#pragma unroll
  for (int v = 0; v < 8; ++v) {
    int r = rbase + v;
    if (r < M) C[(size_t)r * N + n] = acc[v] + bv;
  }
}

// ---------------------------------------------------------------------------
// Per-edge attention score + segment max.
// One wave per edge; lane handles features h=lane and h=lane+32 (coalesced).
// score = dot(leaky_relu(xl[src] + xr[dst] + ea@We, 0.2), att)
// ---------------------------------------------------------------------------
__global__ void edge_score_kernel(const int* __restrict__ ei,
                                  const float* __restrict__ ea,
                                  const float* __restrict__ xl,
                                  const float* __restrict__ xr,
                                  const float* __restrict__ We,
                                  const float* __restrict__ att,
                                  float* __restrict__ score,
                                  float* __restrict__ smax,
                                  int E) {
  __shared__ float sWe[EDIMN * HIDN];
  __shared__ float sAtt[HIDN];
  for (int i = threadIdx.x; i < EDIMN * HIDN; i += blockDim.x) sWe[i] = We[i];
  for (int i = threadIdx.x; i < HIDN; i += blockDim.x) sAtt[i] = att[i];
  __syncthreads();

  const int lane = threadIdx.x & 31;
  const int wid  = threadIdx.x >> 5;
  const int e = blockIdx.x * (blockDim.x >> 5) + wid;
  if (e >= E) return;
  const int s = ei[e];
  const int d = ei[E + e];

  float ebuf[EDIMN];
#pragma unroll
  for (int k = 0; k < EDIMN; ++k) ebuf[k] = ea[(size_t)e * EDIMN + k];

  float partial = 0.0f;
#pragma unroll
  for (int hh = 0; hh < 2; ++hh) {
    const int h = lane + hh * 32;
    float m = xl[(size_t)s * HIDN + h] + xr[(size_t)d * HIDN + h];
#pragma unroll
    for (int k = 0; k < EDIMN; ++k) m += ebuf[k] * sWe[k * HIDN + h];
    const float lr = (m > 0.0f) ? m : 0.2f * m;     // leaky_relu slope 0.2
    partial += lr * sAtt[h];
  }
#pragma unroll
  for (int off = 16; off > 0; off >>= 1) partial += __shfl_xor(partial, off, 32);
  if (lane == 0) {
    score[e] = partial;
    atomicMax(&smax[d], partial);
  }
}

// w = exp(score - smax[dst]);  denom[dst] += w
__global__ void edge_exp_kernel(const int* __restrict__ ei,
                                const float* __restrict__ score,
                                const float* __restrict__ smax,
                                float* __restrict__ w,
                                float* __restrict__ denom, int E) {
  int e = blockIdx.x * blockDim.x + threadIdx.x;
  if (e >= E) return;
  int d = ei[E + e];
  float v = expf(score[e] - smax[d]);
  w[e] = v;
  atomicAdd(&denom[d], v);
}

// agg[dst][h] += (w / (denom[dst]+1e-16)) * xl[src][h]
__global__ void edge_scatter_kernel(const int* __restrict__ ei,
                                    const float* __restrict__ xl,
                                    const float* __restrict__ w,
                                    const float* __restrict__ denom,
                                    float* __restrict__ agg, int E) {
  long long idx = (long long)blockIdx.x * blockDim.x + threadIdx.x;
  int e = (int)(idx >> 6);
  int h = (int)(idx & 63);
  if (e >= E) return;
  int s = ei[e];
  int d = ei[E + e];
  float alpha = w[e] / (denom[d] + 1e-16f);
  atomicAdd(&agg[(size_t)d * HIDN + h], alpha * xl[(size_t)s * HIDN + h]);
}

__global__ void bias_add_kernel(float* __restrict__ agg,
                                const float* __restrict__ b, long long n) {
  long long i = (long long)blockIdx.x * blockDim.x + threadIdx.x;
  if (i < n) agg[i] += b[i & 63];
}

__global__ void fill_kernel(float* __restrict__ p, float v, long long n) {
  long long i = (long long)blockIdx.x * blockDim.x + threadIdx.x;
  if (i < n) p[i] = v;
}

__global__ void relu_kernel(const float* __restrict__ in,
                            float* __restrict__ out, long long n) {
  long long i = (long long)blockIdx.x * blockDim.x + threadIdx.x;
  if (i < n) out[i] = fmaxf(in[i], 0.0f);
}

// ---------------------------------------------------------------------------
extern "C" void kernel_launch(void* const* d_in, const int* in_sizes, int n_in,
                              void* d_out, int out_size, void* d_ws, size_t ws_size,
                              hipStream_t stream) {
  if (n_in < 12) return;
  const int NADR = 100000, NBAT = 100000, NPAR = 50000;
  const int NN[3] = {NADR, NBAT, NPAR};
  const int FF[3] = {32, 64, 16};
  // RELS order: acces(a->b), desservi(b->a), appartient(b->p), contient(p->b)
  const int RS[4] = {0, 1, 1, 2};
  const int RD[4] = {1, 0, 2, 1};
  const int EIX[4] = {3, 5, 7, 9};
  const int EAX[4] = {4, 6, 8, 10};
  const int E = in_sizes[3] / 2;

  const float* X0[3] = {(const float*)d_in[0], (const float*)d_in[1],
                        (const float*)d_in[2]};

  const float *Wl[2][4], *blv[2][4], *Wr[2][4], *brv[2][4], *Wev[2][4],
              *attv[2][4], *bbv[2][4];
  const float *linW[3], *linb[3];

  if (n_in >= 73) {
    // 62 separate param leaves after the 11 data arrays.
    // sorted-key flatten (jax tree_leaves): per-rel leaves {We,Wl,Wr,att,b,bl,br},
    // rels sorted alphabetically; insertion flatten: {Wl,bl,Wr,br,We,att,b}, RELS order.
    const bool sorted = (in_sizes[11] == EDIMN * HIDN);  // We(1024) vs Wl(2048)
    int pos[4];
    if (sorted) { pos[0]=0; pos[1]=3; pos[2]=1; pos[3]=2; }  // acces,appartient,contient,desservi
    else        { pos[0]=0; pos[1]=1; pos[2]=2; pos[3]=3; }
    for (int li = 0; li < 2; ++li)
      for (int r = 0; r < 4; ++r) {
        int base = 11 + li * 28 + pos[r] * 7;
        if (sorted) {
          Wev[li][r]  = (const float*)d_in[base + 0];
          Wl[li][r]   = (const float*)d_in[base + 1];
          Wr[li][r]   = (const float*)d_in[base + 2];
          attv[li][r] = (const float*)d_in[base + 3];
          bbv[li][r]  = (const float*)d_in[base + 4];
          blv[li][r]  = (const float*)d_in[base + 5];
          brv[li][r]  = (const float*)d_in[base + 6];
        } else {
          Wl[li][r]   = (const float*)d_in[base + 0];
          blv[li][r]  = (const float*)d_in[base + 1];
          Wr[li][r]   = (const float*)d_in[base + 2];
          brv[li][r]  = (const float*)d_in[base + 3];
          Wev[li][r]  = (const float*)d_in[base + 4];
          attv[li][r] = (const float*)d_in[base + 5];
          bbv[li][r]  = (const float*)d_in[base + 6];
        }
      }
    for (int t = 0; t < 3; ++t) {
      linW[t] = (const float*)d_in[67 + 2 * t];
      linb[t] = (const float*)d_in[67 + 2 * t + 1];
    }
  } else {
    // single flat params buffer, insertion order
    const float* P = (const float*)d_in[11];
    size_t off = 0;
    for (int li = 0; li < 2; ++li)
      for (int r = 0; r < 4; ++r) {
        int fs = li ? HIDN : FF[RS[r]];
        int fd = li ? HIDN : FF[RD[r]];
        Wl[li][r]   = P + off; off += (size_t)fs * HIDN;
        blv[li][r]  = P + off; off += HIDN;
        Wr[li][r]   = P + off; off += (size_t)fd * HIDN;
        brv[li][r]  = P + off; off += HIDN;
        Wev[li][r]  = P + off; off += EDIMN * HIDN;
        attv[li][r] = P + off; off += HIDN;
        bbv[li][r]  = P + off; off += HIDN;
      }
    for (int t = 0; t < 3; ++t) {
      linW[t] = P + off; off += HIDN * 32;
      linb[t] = P + off; off += 32;
    }
  }

  // workspace layout (floats)
  float* W = (float*)d_ws;
  size_t o = 0;
  float *H[3], *G[3];
  for (int t = 0; t < 3; ++t) { H[t] = W + o; o += (size_t)NN[t] * HIDN; }
  for (int t = 0; t < 3; ++t) { G[t] = W + o; o += (size_t)NN[t] * HIDN; }
  float* XL = W + o; o += (size_t)NADR * HIDN;
  float* XR = W + o; o += (size_t)NADR * HIDN;
  float* SC = W + o; o += (size_t)E;
  float* WB = W + o; o += (size_t)E;
  float* SM = W + o; o += (size_t)NADR;
  float* DN = W + o; o += (size_t)NADR;
  (void)ws_size;

  auto gemm = [&](const float* A, const float* B, const float* bias, float* C,
                  int M, int N, int K) {
    int tiles = ((M + 15) / 16) * (N / 16);
    int blocks = (tiles + 3) / 4;  // 4 waves / 128-thread block
    hipLaunchKernelGGL(gemm_bias_wmma, dim3(blocks), dim3(128), 0, stream,
                       A, B, bias, C, M, N, K);
  };
  auto fill = [&](float* p, float v, long long n) {
    int blocks = (int)((n + 255) / 256);
    hipLaunchKernelGGL(fill_kernel, dim3(blocks), dim3(256), 0, stream, p, v, n);
  };

  const float NEG_INF = -__builtin_inff();

  for (int li = 0; li < 2; ++li) {
    for (int t = 0; t < 3; ++t) fill(G[t], 0.0f, (long long)NN[t] * HIDN);
    for (int r = 0; r < 4; ++r) {
      const int s = RS[r], d = RD[r];
      const int fs = li ? HIDN : FF[s];
      const int fd = li ? HIDN : FF[d];
      const float* Xs = li ? H[s] : X0[s];
      const float* Xd = li ? H[d] : X0[d];
      const int* ei = (const int*)d_in[EIX[r]];
      const float* ea = (const float*)d_in[EAX[r]];

      gemm(Xs, Wl[li][r], blv[li][r], XL, NN[s], HIDN, fs);
      gemm(Xd, Wr[li][r], brv[li][r], XR, NN[d], HIDN, fd);
      fill(SM, NEG_INF, NN[d]);
      fill(DN, 0.0f, NN[d]);
      hipLaunchKernelGGL(edge_score_kernel, dim3((E + 7) / 8), dim3(256), 0,
                         stream, ei, ea, XL, XR, Wev[li][r], attv[li][r],
                         SC, SM, E);
      hipLaunchKernelGGL(edge_exp_kernel, dim3((E + 255) / 256), dim3(256), 0,
                         stream, ei, SC, SM, WB, DN, E);
      long long sc_threads = (long long)E * HIDN;
      hipLaunchKernelGGL(edge_scatter_kernel,
                         dim3((unsigned)((sc_threads + 255) / 256)), dim3(256),
                         0, stream, ei, XL, WB, DN, G[d], E);
      long long bn = (long long)NN[d] * HIDN;
      hipLaunchKernelGGL(bias_add_kernel, dim3((unsigned)((bn + 255) / 256)),
                         dim3(256), 0, stream, G[d], bbv[li][r], bn);
    }
    for (int t = 0; t < 3; ++t) {
      long long n = (long long)NN[t] * HIDN;
      hipLaunchKernelGGL(relu_kernel, dim3((unsigned)((n + 255) / 256)),
                         dim3(256), 0, stream, G[t], H[t], n);
    }
  }

  float* out = (float*)d_out;
  gemm(H[0], linW[0], linb[0], out, NADR, 32, HIDN);
  gemm(H[1], linW[1], linb[1], out + (size_t)NADR * 32, NBAT, 32, HIDN);
  gemm(H[2], linW[2], linb[2], out + (size_t)(NADR + NBAT) * 32, NPAR, 32, HIDN);
  (void)out_size;
}